// SNNL_base_52003464020706
// MI455X (gfx1250) — compile-verified
//
#include <hip/hip_runtime.h>
#include <hip/hip_bf16.h>
#include <math.h>

typedef float v2f __attribute__((ext_vector_type(2)));
typedef float v8f __attribute__((ext_vector_type(8)));

#define BK    32     // K chunk staged in LDS
#define LDA   36     // padded LDS row stride (floats): 144B (16B multiple), bank-conflict-free
#define LOG2E 1.44269504088896341f

// ---------------------------------------------------------------------------
// Kernel 1: per-row reciprocal norms (clamped at 1e-8) + zero top/bot sums.
// ---------------------------------------------------------------------------
__global__ __launch_bounds__(256)
void snnl_prep(const float* __restrict__ x, float* __restrict__ rinv,
               float* __restrict__ top, float* __restrict__ bot, int D)
{
    const int row  = blockIdx.x;
    const int tid  = threadIdx.x;
    const int lane = tid & 31;
    const int wave = tid >> 5;

    const float* xr = x + (size_t)row * D;
    float ss = 0.0f;
    for (int c = tid * 4; c < D; c += 256 * 4) {
        float4 v = *(const float4*)(xr + c);
        ss += v.x * v.x + v.y * v.y + v.z * v.z + v.w * v.w;
    }
    #pragma unroll
    for (int m = 1; m < 32; m <<= 1) ss += __shfl_xor(ss, m);

    __shared__ float ws[8];
    if (lane == 0) ws[wave] = ss;
    __syncthreads();
    if (tid == 0) {
        float t = 0.0f;
        #pragma unroll
        for (int w = 0; w < 8; ++w) t += ws[w];
        float nrm = fmaxf(sqrtf(t), 1e-8f);   // torch cosine_similarity clamp
        rinv[row] = 1.0f / nrm;
        top[row]  = 0.0f;
        bot[row]  = 0.0f;
    }
}

// ---------------------------------------------------------------------------
// Kernel 2: fused symmetric Gram-GEMM + exp + masked row/col reductions.
// 256 threads = 8 wave32, block tile 128x128, wave tile 32x64 (2x4 grid of
// V_WMMA_F32_16X16X4_F32 accumulators). Only colBlock >= rowBlock computed;
// off-diagonal blocks scatter row sums AND column sums (symmetry).
// Global->LDS staging is software-pipelined through registers.
// ---------------------------------------------------------------------------
__global__ __launch_bounds__(256)
void snnl_gemm(const float* __restrict__ x, const int* __restrict__ y,
               const float* __restrict__ T, const float* __restrict__ rinv,
               float* __restrict__ top, float* __restrict__ bot, int D)
{
    const int bx = blockIdx.x;   // column block
    const int by = blockIdx.y;   // row block
    if (bx < by) return;         // block-uniform: EXEC stays all-ones for WMMA

    __shared__ float As[128 * LDA];
    __shared__ float Bs[128 * LDA];
    __shared__ float rR[128], rC[128];
    __shared__ int   yR[128], yC[128];

    const int tid  = threadIdx.x;
    const int lane = tid & 31;
    const int wave = tid >> 5;   // 0..7
    const int wm   = wave & 3;   // wave position in M (0..3), 32 rows each
    const int wn   = wave >> 2;  // wave position in N (0..1), 64 cols each
    const int r    = lane & 15;  // lane within half-wave
    const int h    = lane >> 4;  // half-wave select

    const int rowBase = by * 128;
    const int colBase = bx * 128;

    if (tid < 128) { rR[tid] = rinv[rowBase + tid];       yR[tid] = y[rowBase + tid]; }
    else           { rC[tid-128] = rinv[colBase + tid-128]; yC[tid-128] = y[colBase + tid-128]; }

    const float cc = LOG2E / T[0];   // exp((s-1)/T) = exp2((s-1)*log2e/T)

    // staging geometry: A tile 128x32 = 1024 float4, B tile same; 256 thr x 4 each
    int prRow[4], prC4[4];
    #pragma unroll
    for (int q = 0; q < 4; ++q) {
        const int idx = tid + q * 256;     // 0..1023
        prRow[q] = idx >> 3;
        prC4[q]  = (idx & 7) << 2;
    }

    v8f acc[2][4] = {};
    float4 pA[4], pB[4];

    // prologue prefetch of chunk 0
    #pragma unroll
    for (int q = 0; q < 4; ++q) {
        pA[q] = *(const float4*)(x + (size_t)(rowBase + prRow[q]) * D + prC4[q]);
        pB[q] = *(const float4*)(x + (size_t)(colBase + prRow[q]) * D + prC4[q]);
    }

    for (int k0 = 0; k0 < D; k0 += BK) {
        __syncthreads();   // previous chunk's fragment reads finished
        // regs -> LDS, normalizing on the fly
        #pragma unroll
        for (int q = 0; q < 4; ++q) {
            const float rv = rR[prRow[q]];
            *(float4*)&As[prRow[q] * LDA + prC4[q]] =
                make_float4(pA[q].x * rv, pA[q].y * rv, pA[q].z * rv, pA[q].w * rv);
            const float cv = rC[prRow[q]];
            *(float4*)&Bs[prRow[q] * LDA + prC4[q]] =
                make_float4(pB[q].x * cv, pB[q].y * cv, pB[q].z * cv, pB[q].w * cv);
        }
        // issue prefetch of next chunk; latency hidden behind this chunk's WMMAs
        const int kn = k0 + BK;
        if (kn < D) {
            #pragma unroll
            for (int q = 0; q < 4; ++q) {
                pA[q] = *(const float4*)(x + (size_t)(rowBase + prRow[q]) * D + kn + prC4[q]);
                pB[q] = *(const float4*)(x + (size_t)(colBase + prRow[q]) * D + kn + prC4[q]);
            }
        }
        __syncthreads();

        // ISA f32 16x4 A/B layout: lanes 0-15 -> K={kk,kk+1}, lanes 16-31 -> K={kk+2,kk+3}
        #pragma unroll
        for (int kk = 0; kk < BK; kk += 4) {
            const int ko = kk + 2 * h;
            v2f a[2], b[4];
            #pragma unroll
            for (int im = 0; im < 2; ++im)
                a[im] = *(const v2f*)&As[(wm * 32 + im * 16 + r) * LDA + ko];
            #pragma unroll
            for (int in = 0; in < 4; ++in)
                b[in] = *(const v2f*)&Bs[(wn * 64 + in * 16 + r) * LDA + ko];
            #pragma unroll
            for (int im = 0; im < 2; ++im)
                #pragma unroll
                for (int in = 0; in < 4; ++in)
                    acc[im][in] = __builtin_amdgcn_wmma_f32_16x16x4_f32(
                        false, a[im], false, b[in], (short)0, acc[im][in], false, false);
        }
    }

    // -------- epilogue: f = exp2((sim-1)*cc), masked reductions --------
    // C/D layout: VGPR v, lanes 0-15 -> (M = v, N = lane); lanes 16-31 -> (M = v+8, N = lane-16)
    float cbot[4] = {0.f, 0.f, 0.f, 0.f};
    float ctop[4] = {0.f, 0.f, 0.f, 0.f};
    const bool isDiag = (bx == by);

    #pragma unroll
    for (int im = 0; im < 2; ++im) {
        #pragma unroll
        for (int v = 0; v < 8; ++v) {
            const int mloc = wm * 32 + im * 16 + v + 8 * h;
            const int mg   = rowBase + mloc;
            const int yr   = yR[mloc];
            float fb = 0.0f, ft = 0.0f;
            #pragma unroll
            for (int in = 0; in < 4; ++in) {
                const int nloc = wn * 64 + in * 16 + r;
                const int jg   = colBase + nloc;
                float s = acc[im][in][v];
                float f = exp2f((s - 1.0f) * cc);
                if (mg == jg) f = 0.0f;                       // exclude i == j
                const float fsame = (yr == yC[nloc]) ? f : 0.0f;
                fb += f;  ft += fsame;
                cbot[in] += f;  ctop[in] += fsame;            // column accumulators
            }
            // row sum over 64 columns: per-lane partials then 16-lane xor reduce
            #pragma unroll
            for (int mm = 1; mm < 16; mm <<= 1) {
                fb += __shfl_xor(fb, mm);
                ft += __shfl_xor(ft, mm);
            }
            if (r == 0) {          // lanes 0 (h=0) and 16 (h=1): rows v and v+8
                atomicAdd(&bot[mg], fb);
                atomicAdd(&top[mg], ft);
            }
        }
    }

    // symmetric scatter: column sums -> top/bot[j] (skip on diagonal blocks)
    if (!isDiag) {
        #pragma unroll
        for (int in = 0; in < 4; ++in) {
            float cb = cbot[in] + __shfl_xor(cbot[in], 16);
            float ct = ctop[in] + __shfl_xor(ctop[in], 16);
            if (h == 0) {
                const int jg = colBase + wn * 64 + in * 16 + r;
                atomicAdd(&bot[jg], cb);
                atomicAdd(&top[jg], ct);
            }
        }
    }
}

// ---------------------------------------------------------------------------
// Kernel 3: loss = mean(log(bot) - log(top + 1e-9))
// ---------------------------------------------------------------------------
__global__ __launch_bounds__(256)
void snnl_final(const float* __restrict__ top, const float* __restrict__ bot,
                float* __restrict__ out, int N)
{
    const int tid  = threadIdx.x;
    const int lane = tid & 31;
    const int wave = tid >> 5;

    float s = 0.0f;
    for (int i = tid; i < N; i += 256)
        s += logf(bot[i]) - logf(top[i] + 1e-9f);

    #pragma unroll
    for (int m = 1; m < 32; m <<= 1) s += __shfl_xor(s, m);

    __shared__ float ws[8];
    if (lane == 0) ws[wave] = s;
    __syncthreads();
    if (tid == 0) {
        float t = 0.0f;
        #pragma unroll
        for (int w = 0; w < 8; ++w) t += ws[w];
        out[0] = t / (float)N;
    }
}

// ---------------------------------------------------------------------------
extern "C" void kernel_launch(void* const* d_in, const int* in_sizes, int n_in,
                              void* d_out, int out_size, void* d_ws, size_t ws_size,
                              hipStream_t stream)
{
    const float* x = (const float*)d_in[0];
    const int*   y = (const int*)d_in[1];
    const float* T = (const float*)d_in[2];
    float* out = (float*)d_out;

    const int N = in_sizes[1];            // 8192
    const int D = in_sizes[0] / N;        // 1024

    float* rinv = (float*)d_ws;           // [N]
    float* top  = rinv + N;               // [N]
    float* bot  = top + N;                // [N]  (total 96 KB of scratch)

    snnl_prep<<<N, 256, 0, stream>>>(x, rinv, top, bot, D);

    dim3 grid(N / 128, N / 128);          // upper-triangle blocks do the work
    snnl_gemm<<<grid, 256, 0, stream>>>(x, y, T, rinv, top, bot, D);

    snnl_final<<<1, 256, 0, stream>>>(top, bot, out, N);
}